// NLTKHierarchicalSoftmax_80968723464281
// MI455X (gfx1250) — compile-verified
//
#include <hip/hip_runtime.h>

// Problem constants (match reference)
#define BATCH 4096
#define NHID  512
#define BR    32

typedef __attribute__((ext_vector_type(2))) float v2f;
typedef __attribute__((ext_vector_type(8))) float v8f;

// ---------------------------------------------------------------------------
// Kernel 1: level-0 logits = inputs(4096x512) @ W[node 0](512x32), fp32 WMMA.
// One wave per 16-batch tile; each wave produces a 16x32 tile using two
// V_WMMA_F32_16X16X4_F32 accumulators (N columns 0..15 and 16..31), K-loop
// over 512 in steps of 4.
//
// ISA operand layouts (CDNA5 ISA):
//   A (16x4 f32, 2 VGPRs):  lane l -> p=l>>4, r=l&15 holds A[M=r, K=2p], A[M=r, K=2p+1]
//   B (4x16 f32, 2 VGPRs):  lane l holds B[K=2p, N=r], B[K=2p+1, N=r]
//   D (16x16 f32, 8 VGPRs): vgpr v, lane l -> D[M = v + 8p, N = r]
// ---------------------------------------------------------------------------
__global__ __launch_bounds__(128) void hs_level0_wmma(
    const float* __restrict__ inp,   // (4096, 512)
    const float* __restrict__ W,     // (1057, 512, 32); node 0 at offset 0
    float* __restrict__ logits0)     // (4096, 32) scratch
{
  const int wave = threadIdx.x >> 5;
  const int lane = threadIdx.x & 31;
  const int tile = blockIdx.x * 4 + wave;   // 256 tiles of 16 batches
  const int r = lane & 15;
  const int p = lane >> 4;

  const float* arow = inp + (size_t)(tile * 16 + r) * NHID;

  v8f c0 = {};  // columns 0..15
  v8f c1 = {};  // columns 16..31

  #pragma unroll 4
  for (int kb = 0; kb < NHID; kb += 4) {
    const int k = kb + 2 * p;
    v2f a, b0, b1;
    a[0]  = arow[k];
    a[1]  = arow[k + 1];
    b0[0] = W[k * BR + r];
    b0[1] = W[(k + 1) * BR + r];
    b1[0] = W[k * BR + 16 + r];
    b1[1] = W[(k + 1) * BR + 16 + r];
    c0 = __builtin_amdgcn_wmma_f32_16x16x4_f32(false, a, false, b0,
                                               (short)0, c0, false, false);
    c1 = __builtin_amdgcn_wmma_f32_16x16x4_f32(false, a, false, b1,
                                               (short)0, c1, false, false);
  }

  // Store D: lane (p, r), vgpr v -> logits0[tile*16 + v + 8p][r] and [r+16]
  float* outbase = logits0 + (size_t)(tile * 16 + 8 * p) * BR + r;
  #pragma unroll
  for (int v = 0; v < 8; ++v) {
    outbase[v * BR]      = c0[v];
    outbase[v * BR + 16] = c1[v];
  }
}

// ---------------------------------------------------------------------------
// Kernel 2: per-batch wave. Lane = output column c (0..31).
// Computes level-1/2 matvecs (coalesced: lane c reads W[node, h, c], 128 B
// contiguous per h-row across the wave), then 3 softmaxes over 32 lanes via
// wave32 shuffles, picks the step probability per level, writes the product.
// ---------------------------------------------------------------------------
__global__ __launch_bounds__(128) void hs_finish(
    const float* __restrict__ inp,      // (4096, 512)
    const int*   __restrict__ labels,   // (4096,)
    const float* __restrict__ W,        // (1057, 512, 32)
    const float* __restrict__ logits0,  // (4096, 32) from kernel 1
    float* __restrict__ out)            // (4096,)
{
  const int wave = threadIdx.x >> 5;
  const int lane = threadIdx.x & 31;
  const int b = blockIdx.x * 4 + wave;

  const int label = labels[b];
  const int n1 = 1  + (label >> 10);        // level-1 node
  const int n2 = 33 + (label >> 5);         // level-2 node
  int steps[3];
  steps[0] = (label >> 10) & 31;
  steps[1] = (label >> 5) & 31;
  steps[2] = label & 31;

  const float* x  = inp + (size_t)b * NHID;
  const float* w1 = W + (size_t)n1 * NHID * BR + lane;
  const float* w2 = W + (size_t)n2 * NHID * BR + lane;

  float acc1 = 0.f, acc2 = 0.f;
  #pragma unroll 8
  for (int h = 0; h < NHID; ++h) {
    const float xv = x[h];                  // wave-uniform broadcast
    acc1 = fmaf(xv, w1[(size_t)h * BR], acc1);
    acc2 = fmaf(xv, w2[(size_t)h * BR], acc2);
  }

  float logit[3];
  logit[0] = logits0[(size_t)b * BR + lane];
  logit[1] = acc1;
  logit[2] = acc2;

  float prod = 1.f;
  #pragma unroll
  for (int d = 0; d < 3; ++d) {
    // wave32 max-reduce
    float m = logit[d];
    #pragma unroll
    for (int off = 16; off > 0; off >>= 1)
      m = fmaxf(m, __shfl_xor(m, off, 32));
    const float e = __expf(logit[d] - m);
    // wave32 sum-reduce
    float s = e;
    #pragma unroll
    for (int off = 16; off > 0; off >>= 1)
      s += __shfl_xor(s, off, 32);
    const float pd = e / s;
    prod *= __shfl(pd, steps[d], 32);       // step index is wave-uniform
  }

  if (lane == 0) out[b] = prod;
}

extern "C" void kernel_launch(void* const* d_in, const int* in_sizes, int n_in,
                              void* d_out, int out_size, void* d_ws, size_t ws_size,
                              hipStream_t stream) {
  const float* inp    = (const float*)d_in[0];   // (4096, 512) f32
  const int*   labels = (const int*)d_in[1];     // (4096,) i32
  const float* W      = (const float*)d_in[2];   // (1057, 512, 32) f32
  float* out = (float*)d_out;                    // (4096,) f32
  float* logits0 = (float*)d_ws;                 // needs 4096*32*4 = 512 KB

  // 256 batch tiles of 16, 4 waves (128 threads) per block -> 64 blocks
  hs_level0_wmma<<<64, 128, 0, stream>>>(inp, W, logits0);
  // 4096 batches, 4 waves per block -> 1024 blocks
  hs_finish<<<1024, 128, 0, stream>>>(inp, labels, W, logits0, out);
}